// KWNG_73847667687658
// MI455X (gfx1250) — compile-verified
//
#include <hip/hip_runtime.h>
#include <math.h>

// ---------------------------------------------------------------------------
// KWNG natural gradient for MI455X (gfx1250, wave32).
// Bandwidth-bound (~832 MB of traffic, ~1.5 GFLOP): two streaming passes over
// T/g, tiny 5x5 algebra on single threads, V_WMMA_F32_16X16X4_F32 for the
// Tt = W @ T (5x5 x 5xD) GEMM so the VALU is free for the per-column work.
// Dv uses v_sqrt_f32/v_rcp_f32 hardware approximations (~1 ulp) to keep the
// per-column VALU cost ~10x below the HBM roofline (23.3 TB/s -> ~36 us).
// ---------------------------------------------------------------------------

typedef __attribute__((ext_vector_type(2))) float v2f;
typedef __attribute__((ext_vector_type(8))) float v8f;

#define NUM_BASIS 5
#define EPS_C     1e-5f
#define BETA_C    (-1.0f)
#define THRESH_C  0.0f

#define NB        2048      // blocks for streaming passes (8 waves each)
#define NTHR      256

// workspace layout (floats)
#define WS_W      0         // 25: whitening matrix W = diag(ssinv) @ V^T
#define WS_MASK   32        // 5 : kept-singular-direction mask (as float)
#define WS_X      40        // 5 : solution of G x = cond_g
#define WS_PART   64        // NB*20 : per-block partials (cg[5], G15[15])

// ---------------------------------------------------------------------------
// Kernel 1: 5x5 symmetric PSD eigendecomposition (cyclic Jacobi) -> W, mask.
// ---------------------------------------------------------------------------
__global__ void kwng_setup(const float* __restrict__ Kin, float* __restrict__ ws) {
  if (blockIdx.x != 0 || threadIdx.x != 0) return;
  float A[5][5], V[5][5];
  for (int i = 0; i < 5; ++i)
    for (int j = 0; j < 5; ++j) {
      A[i][j] = Kin[i * 5 + j];
      V[i][j] = (i == j) ? 1.0f : 0.0f;
    }
  for (int sweep = 0; sweep < 30; ++sweep) {
    float off = 0.0f;
    for (int i = 0; i < 5; ++i)
      for (int j = i + 1; j < 5; ++j) off += A[i][j] * A[i][j];
    if (off < 1e-18f) break;
    for (int p = 0; p < 4; ++p) {
      for (int q = p + 1; q < 5; ++q) {
        float apq = A[p][q];
        if (fabsf(apq) < 1e-30f) continue;
        float app = A[p][p], aqq = A[q][q];
        float theta = 0.5f * (aqq - app) / apq;
        float t = copysignf(1.0f, theta) / (fabsf(theta) + sqrtf(theta * theta + 1.0f));
        float c = 1.0f / sqrtf(t * t + 1.0f);
        float s = t * c;
        for (int k = 0; k < 5; ++k) {  // A <- A J  (columns p,q)
          float akp = A[k][p], akq = A[k][q];
          A[k][p] = c * akp - s * akq;
          A[k][q] = s * akp + c * akq;
        }
        for (int k = 0; k < 5; ++k) {  // A <- J^T A (rows p,q)
          float apk = A[p][k], aqk = A[q][k];
          A[p][k] = c * apk - s * aqk;
          A[q][k] = s * apk + c * aqk;
        }
        for (int k = 0; k < 5; ++k) {  // V <- V J (eigenvectors are columns)
          float vkp = V[k][p], vkq = V[k][q];
          V[k][p] = c * vkp - s * vkq;
          V[k][q] = s * vkp + c * vkq;
        }
      }
    }
  }
  // SVD of symmetric PSD: singular values = |eigenvalues|, vh rows = eigvecs^T.
  for (int i = 0; i < 5; ++i) {
    float sv = fabsf(A[i][i]);
    float mk = (sv > THRESH_C) ? 1.0f : 0.0f;
    float ssinv = (sv > THRESH_C) ? (1.0f / sqrtf(sv)) : 0.0f;  // sqrt(1/s)
    ws[WS_MASK + i] = mk;
    for (int j = 0; j < 5; ++j) ws[WS_W + i * 5 + j] = ssinv * V[j][i];
  }
}

// ---------------------------------------------------------------------------
// Shared tile engine: one wave computes Tt[0..4][16 cols] via 2x
// V_WMMA_F32_16X16X4_F32 (K=5 split across slices k=0..3 and k=4..7, padded).
// A layout (16x4 f32): lane M=L%16; VGPR0 holds K=2h, VGPR1 K=1+2h (h=L/16).
// B layout (4x16 f32): lane N=L%16; VGPR0 holds K=2h, VGPR1 K=1+2h.
// D layout: VGPR r holds row M=r+8h, col N=L%16. Rows 5..15 of padded W are
// zero, so lanes with h==1 get exact zeros in c[0..4] -> inert for reductions.
// ---------------------------------------------------------------------------
__device__ __forceinline__ void load_a_frags(const float* __restrict__ W,
                                             int n, int h, v2f& a0, v2f& a1) {
  bool rowok = (n < NUM_BASIS);
  a0.x = rowok ? W[n * 5 + 2 * h] : 0.0f;      // k = 0 or 2
  a0.y = rowok ? W[n * 5 + 2 * h + 1] : 0.0f;  // k = 1 or 3
  a1.x = (rowok && h == 0) ? W[n * 5 + 4] : 0.0f;  // k = 4 (k=6 pad -> 0)
  a1.y = 0.0f;                                      // k = 5 or 7 pad
}

__device__ __forceinline__ v8f tt_tile(const float* __restrict__ T, long long D,
                                       long long col, int h, v2f a0, v2f a1) {
  v2f b0, b1;
  b0.x = T[(long long)(2 * h) * D + col];        // rows 0 / 2
  b0.y = T[(long long)(2 * h + 1) * D + col];    // rows 1 / 3
  float r4 = T[4ll * D + col];                   // both halves load same 64B
  b1.x = (h == 0) ? r4 : 0.0f;                   // row 4 (row 6 pad -> 0)
  b1.y = 0.0f;                                   // rows 5 / 7 pad
  v8f c = {};
  c = __builtin_amdgcn_wmma_f32_16x16x4_f32(false, a0, false, b0, (short)0, c,
                                            false, false);
  c = __builtin_amdgcn_wmma_f32_16x16x4_f32(false, a1, false, b1, (short)0, c,
                                            false, false);
  return c;
}

// Dv = 1 / (sqrt(q) + 1e-8): hardware v_sqrt_f32 + v_rcp_f32 approximations
// (~1 ulp) instead of the ~45-instruction IEEE sqrt+divide expansion.
__device__ __forceinline__ float dv_of(float q) {
  return __builtin_amdgcn_rcpf(__builtin_amdgcn_sqrtf(q) + 1e-8f);
}

// ---------------------------------------------------------------------------
// Kernel 2 (pass 1): stream T,g; accumulate cond_g[5] and upper-tri G[15].
// Deterministic: shfl_xor wave reduce -> LDS block combine -> fixed slot.
// ---------------------------------------------------------------------------
__global__ void kwng_pass1(const float* __restrict__ T, const float* __restrict__ g,
                           float* __restrict__ ws, int Dn) {
  const long long D = (long long)Dn;
  const int lane = threadIdx.x & 31;
  const int wave = threadIdx.x >> 5;
  const int n = lane & 15, h = lane >> 4;

  v2f a0, a1;
  load_a_frags(ws + WS_W, n, h, a0, a1);

  float acc[20];
#pragma unroll
  for (int v = 0; v < 20; ++v) acc[v] = 0.0f;

  const int tiles = Dn >> 4;
  const int gw = blockIdx.x * (blockDim.x >> 5) + wave;
  const int nw = gridDim.x * (blockDim.x >> 5);

#pragma unroll 2
  for (int t = gw; t < tiles; t += nw) {
    const long long col = ((long long)t << 4) + n;
    float gv = g[col];  // h==1 lanes: tt==0 -> contributions are 0 anyway
    v8f c = tt_tile(T, D, col, h, a0, a1);
    float tt[5];
#pragma unroll
    for (int m = 0; m < 5; ++m) tt[m] = c[m];
    float q = tt[0] * tt[0] + tt[1] * tt[1] + tt[2] * tt[2] +
              tt[3] * tt[3] + tt[4] * tt[4];
    float dv = dv_of(q);
    float dg = dv * gv;
    int idx = 5;
#pragma unroll
    for (int m = 0; m < 5; ++m) {
      acc[m] += tt[m] * dg;          // cond_g
      float sm = tt[m] * dv;
#pragma unroll
      for (int nn = m; nn < 5; ++nn) acc[idx++] += sm * tt[nn];  // G upper-tri
    }
  }

  __shared__ float lds[NTHR / 32][20];
#pragma unroll
  for (int v = 0; v < 20; ++v) {
    float x = acc[v];
    for (int m = 16; m; m >>= 1) x += __shfl_xor(x, m, 32);
    acc[v] = x;
  }
  if (lane == 0)
#pragma unroll
    for (int v = 0; v < 20; ++v) lds[wave][v] = acc[v];
  __syncthreads();
  if (threadIdx.x < 20) {
    float s = 0.0f;
    for (int w = 0; w < (int)(blockDim.x >> 5); ++w) s += lds[w][threadIdx.x];
    ws[WS_PART + blockIdx.x * 20 + threadIdx.x] = s;
  }
}

// ---------------------------------------------------------------------------
// Kernel 3: fixed-order reduction of NB partials, then 5x5 SPD solve.
// ---------------------------------------------------------------------------
__global__ void kwng_solve(float* __restrict__ ws) {
  __shared__ float red[NTHR];
  __shared__ float tot[20];
  const float* P = ws + WS_PART;
  const int tid = threadIdx.x;
  for (int v = 0; v < 20; ++v) {
    float s = 0.0f;
    for (int i = tid; i < NB; i += NTHR) s += P[i * 20 + v];
    red[tid] = s;
    __syncthreads();
    for (int w = NTHR / 2; w; w >>= 1) {
      if (tid < w) red[tid] += red[tid + w];
      __syncthreads();
    }
    if (tid == 0) tot[v] = red[0];
    __syncthreads();
  }
  if (tid == 0) {
    float G[5][5], b[5], x[5];
    for (int m = 0; m < 5; ++m) b[m] = tot[m];
    int idx = 5;
    for (int m = 0; m < 5; ++m)
      for (int nn = m; nn < 5; ++nn) {
        float v = tot[idx++];
        G[m][nn] = v;
        G[nn][m] = v;
      }
    for (int m = 0; m < 5; ++m) G[m][m] += EPS_C * ws[WS_MASK + m];
    // Gaussian elimination with partial pivoting
    for (int col = 0; col < 5; ++col) {
      int pr = col;
      float best = fabsf(G[col][col]);
      for (int r = col + 1; r < 5; ++r)
        if (fabsf(G[r][col]) > best) { best = fabsf(G[r][col]); pr = r; }
      if (pr != col) {
        for (int k = 0; k < 5; ++k) { float tmp = G[col][k]; G[col][k] = G[pr][k]; G[pr][k] = tmp; }
        float tb = b[col]; b[col] = b[pr]; b[pr] = tb;
      }
      float inv = 1.0f / G[col][col];
      for (int r = col + 1; r < 5; ++r) {
        float f = G[r][col] * inv;
        for (int k = col; k < 5; ++k) G[r][k] -= f * G[col][k];
        b[r] -= f * b[col];
      }
    }
    for (int r = 4; r >= 0; --r) {
      float s = b[r];
      for (int k = r + 1; k < 5; ++k) s -= G[r][k] * x[k];
      x[r] = s / G[r][r];
    }
    for (int i = 0; i < 5; ++i) ws[WS_X + i] = x[i];
  }
}

// ---------------------------------------------------------------------------
// Kernel 4 (pass 2): recompute Tt tile + Dv (cheaper than materializing 320MB),
// emit out = Dv * (g - BETA * Tt^T x) / EPS.
// ---------------------------------------------------------------------------
__global__ void kwng_pass2(const float* __restrict__ T, const float* __restrict__ g,
                           const float* __restrict__ ws, float* __restrict__ out,
                           int Dn) {
  const long long D = (long long)Dn;
  const int lane = threadIdx.x & 31;
  const int wave = threadIdx.x >> 5;
  const int n = lane & 15, h = lane >> 4;

  v2f a0, a1;
  load_a_frags(ws + WS_W, n, h, a0, a1);
  float x0 = ws[WS_X + 0], x1 = ws[WS_X + 1], x2 = ws[WS_X + 2],
        x3 = ws[WS_X + 3], x4 = ws[WS_X + 4];

  const int tiles = Dn >> 4;
  const int gw = blockIdx.x * (blockDim.x >> 5) + wave;
  const int nw = gridDim.x * (blockDim.x >> 5);

#pragma unroll 2
  for (int t = gw; t < tiles; t += nw) {
    const long long col = ((long long)t << 4) + n;
    float gv = g[col];
    v8f c = tt_tile(T, D, col, h, a0, a1);
    float q = c[0] * c[0] + c[1] * c[1] + c[2] * c[2] + c[3] * c[3] + c[4] * c[4];
    float dv = dv_of(q);
    float dot = c[0] * x0 + c[1] * x1 + c[2] * x2 + c[3] * x3 + c[4] * x4;
    float res = dv * (gv - BETA_C * dot) * (1.0f / EPS_C);
    if (h == 0) out[col] = res;  // lanes 16-31 hold padded rows -> no store
  }
  (void)wave;
}

// ---------------------------------------------------------------------------
extern "C" void kernel_launch(void* const* d_in, const int* in_sizes, int n_in,
                              void* d_out, int out_size, void* d_ws, size_t ws_size,
                              hipStream_t stream) {
  const float* K = (const float*)d_in[0];
  const float* T = (const float*)d_in[1];
  const float* g = (const float*)d_in[2];
  float* out = (float*)d_out;
  float* ws = (float*)d_ws;
  const int D = in_sizes[2];

  kwng_setup<<<1, 32, 0, stream>>>(K, ws);
  kwng_pass1<<<NB, NTHR, 0, stream>>>(T, g, ws, D);
  kwng_solve<<<1, NTHR, 0, stream>>>(ws);
  kwng_pass2<<<NB, NTHR, 0, stream>>>(T, g, ws, out, D);
  (void)n_in; (void)out_size; (void)ws_size;
}